// AntiSymmetricConv_14310831030632
// MI455X (gfx1250) — compile-verified
//
#include <hip/hip_runtime.h>
#include <hip/hip_bf16.h>
#include <math.h>

#define D 128
#define TILE_M 16
#define GAMMA 0.1f
#define EPSILON 0.1f
#define NUM_ITERS 4

typedef __attribute__((ext_vector_type(2))) float v2f;
typedef __attribute__((ext_vector_type(8))) float v8f;

// ---------------------------------------------------------------------------
// Degree / dinv precompute (edge_index is constant across iterations)
// ---------------------------------------------------------------------------
__global__ __launch_bounds__(256) void init_deg_kernel(float* __restrict__ deg, int n) {
    int i = blockIdx.x * blockDim.x + threadIdx.x;
    if (i < n) deg[i] = 1.0f;  // self-loop
}

__global__ __launch_bounds__(256) void count_deg_kernel(const int* __restrict__ ei,
                                                        float* __restrict__ deg, int E) {
    int e = blockIdx.x * blockDim.x + threadIdx.x;
    if (e < E) {
        int dst = ei[E + e];
        __hip_atomic_fetch_add(&deg[dst], 1.0f, __ATOMIC_RELAXED, __HIP_MEMORY_SCOPE_AGENT);
    }
}

__global__ __launch_bounds__(256) void finalize_dinv_kernel(float* __restrict__ deg, int n) {
    int i = blockIdx.x * blockDim.x + threadIdx.x;
    if (i < n) deg[i] = rsqrtf(deg[i]);  // in place: deg -> dinv
}

// ---------------------------------------------------------------------------
// Weight prep into K-pair-packed transposed layout:
//   T[k][n]  packed at  [(k>>1)*2D + n*2 + (k&1)]
// so a WMMA B-fragment (B[k][n], B[k+1][n]) is one aligned 8B load, and
// lanes 0..15 (consecutive n) cover a contiguous 128B segment.
//   WphiT[k][n] = W_phi[n][k]
//   AT[k][n]    = A[n][k] = W[n][k] - W[k][n] - gamma*(k==n)
// ---------------------------------------------------------------------------
__global__ __launch_bounds__(256) void prep_weights_kernel(const float* __restrict__ W,
                                                           const float* __restrict__ Wphi,
                                                           float* __restrict__ WphiP,
                                                           float* __restrict__ ATP) {
    int i = blockIdx.x * blockDim.x + threadIdx.x;  // i = k*D + n
    if (i >= D * D) return;
    int k = i / D, n = i % D;
    size_t po = (size_t)(k >> 1) * (2 * D) + n * 2 + (k & 1);
    WphiP[po] = Wphi[n * D + k];
    float a = W[n * D + k] - W[k * D + n];
    if (k == n) a -= GAMMA;
    ATP[po] = a;
}

// ---------------------------------------------------------------------------
// Fused dual GEMM (f32 WMMA 16x16x4, exact f32 like the reference):
//   xw  = x @ Wphi^T
//   agg = x @ A^T + bias          (edge scatter accumulates on top of agg)
// One block = one 16-row x-tile in LDS; 8 waves = 8 column tiles; each wave
// produces both outputs for its 16-column slab, sharing the A fragment.
// ---------------------------------------------------------------------------
__global__ __launch_bounds__(256) void gemm_dual_kernel(const float* __restrict__ x,
                                                        const float* __restrict__ WphiP,
                                                        const float* __restrict__ ATP,
                                                        const float* __restrict__ bias,
                                                        float* __restrict__ xw,
                                                        float* __restrict__ agg, int N) {
    __shared__ float lds_x[TILE_M][D + 4];  // +4 pad: conflict-free fragment reads

    const int row0 = blockIdx.x * TILE_M;
    const bool full = (row0 + TILE_M) <= N;
    const int tid = threadIdx.x;

    // Stage the 16x128 f32 x-tile with coalesced float4 loads (2048 floats).
    for (int i = tid; i < TILE_M * (D / 4); i += 256) {
        int r = i / (D / 4);
        int c = (i % (D / 4)) * 4;
        float4 v = make_float4(0.f, 0.f, 0.f, 0.f);
        if (full || (row0 + r < N)) v = *(const float4*)(x + (size_t)(row0 + r) * D + c);
        lds_x[r][c + 0] = v.x; lds_x[r][c + 1] = v.y;
        lds_x[r][c + 2] = v.z; lds_x[r][c + 3] = v.w;
    }
    __syncthreads();

    const int wave = tid >> 5;         // 0..7 -> column tile
    const int lane = tid & 31;
    const int mn   = lane & 15;        // M (A frag) / N (B,C frags)
    const int kq   = lane >> 4;        // lanes 16-31 hold K+2,K+3 (pair index +1)
    const int col0 = wave * 16;
    const size_t bcol = (size_t)(col0 + mn) * 2;

    v8f acc_w = {};
    v8f acc_a = {};

#pragma unroll 4
    for (int k0 = 0; k0 < D; k0 += 4) {
        const int k = k0 + kq * 2;
        const int p = (k0 >> 1) + kq;  // K-pair index for packed weights
        const v2f a  = *(const v2f*)&lds_x[mn][k];
        const v2f bw = *(const v2f*)(WphiP + (size_t)p * (2 * D) + bcol);
        const v2f ba = *(const v2f*)(ATP   + (size_t)p * (2 * D) + bcol);
        acc_w = __builtin_amdgcn_wmma_f32_16x16x4_f32(false, a, false, bw,
                                                      (short)0, acc_w, false, false);
        acc_a = __builtin_amdgcn_wmma_f32_16x16x4_f32(false, a, false, ba,
                                                      (short)0, acc_a, false, false);
    }

    const float b    = bias[col0 + mn];
    const int rowoff = kq * 8;  // C/D layout: vgpr r -> row r (lanes>=16: +8)
    if (full) {
#pragma unroll
        for (int r = 0; r < 8; ++r) {
            const size_t o = (size_t)(row0 + rowoff + r) * D + col0 + mn;
            xw[o]  = acc_w[r];
            agg[o] = acc_a[r] + b;
        }
    } else {
#pragma unroll
        for (int r = 0; r < 8; ++r) {
            const int row = row0 + rowoff + r;
            if (row < N) {
                const size_t o = (size_t)row * D + col0 + mn;
                xw[o]  = acc_w[r];
                agg[o] = acc_a[r] + b;
            }
        }
    }
}

// ---------------------------------------------------------------------------
// Edge scatter: one wave per edge; lane handles 16B of the 512B row.
//   agg[dst] += xw[src] * dinv[src]*dinv[dst]
// Gather is a full coalesced 512B cacheline; adds are hardware f32 L2 atomics.
// ---------------------------------------------------------------------------
__global__ __launch_bounds__(256) void edge_scatter_kernel(const int* __restrict__ ei,
                                                           const float* __restrict__ xw,
                                                           const float* __restrict__ dinv,
                                                           float* __restrict__ agg, int E) {
    const int wid  = (int)((blockIdx.x * (size_t)blockDim.x + threadIdx.x) >> 5);
    const int lane = threadIdx.x & 31;
    if (wid >= E) return;
    const int s = ei[wid];
    const int d = ei[E + wid];
    const float norm = dinv[s] * dinv[d];
    const float4 v = *(const float4*)(xw + (size_t)s * D + lane * 4);
    float* p = agg + (size_t)d * D + lane * 4;
    __hip_atomic_fetch_add(p + 0, v.x * norm, __ATOMIC_RELAXED, __HIP_MEMORY_SCOPE_AGENT);
    __hip_atomic_fetch_add(p + 1, v.y * norm, __ATOMIC_RELAXED, __HIP_MEMORY_SCOPE_AGENT);
    __hip_atomic_fetch_add(p + 2, v.z * norm, __ATOMIC_RELAXED, __HIP_MEMORY_SCOPE_AGENT);
    __hip_atomic_fetch_add(p + 3, v.w * norm, __ATOMIC_RELAXED, __HIP_MEMORY_SCOPE_AGENT);
}

// ---------------------------------------------------------------------------
// Combine: x_out = x_in + eps * tanh(agg + xw * dinv^2)  (elementwise, in-place safe)
// ---------------------------------------------------------------------------
__global__ __launch_bounds__(256) void combine_kernel(const float* __restrict__ xin,
                                                      const float* __restrict__ xw,
                                                      const float* __restrict__ agg,
                                                      const float* __restrict__ dinv,
                                                      float* __restrict__ xout, int N) {
    const int i = blockIdx.x * blockDim.x + threadIdx.x;  // over N*D/4 float4s
    const int total = N * (D / 4);
    if (i >= total) return;
    const int node = i / (D / 4);
    const float di = dinv[node];
    const float d2 = di * di;
    const size_t o = (size_t)i * 4;
    const float4 xv = *(const float4*)(xin + o);
    const float4 wv = *(const float4*)(xw + o);
    const float4 av = *(const float4*)(agg + o);
    float4 r;
    r.x = xv.x + EPSILON * tanhf(av.x + wv.x * d2);
    r.y = xv.y + EPSILON * tanhf(av.y + wv.y * d2);
    r.z = xv.z + EPSILON * tanhf(av.z + wv.z * d2);
    r.w = xv.w + EPSILON * tanhf(av.w + wv.w * d2);
    *(float4*)(xout + o) = r;
}

// ---------------------------------------------------------------------------
// Launcher
// inputs: x[N*D] f32, edge_index[2*E] i32, W[D*D] f32, W_phi[D*D] f32, bias[D] f32
// ---------------------------------------------------------------------------
extern "C" void kernel_launch(void* const* d_in, const int* in_sizes, int n_in,
                              void* d_out, int out_size, void* d_ws, size_t ws_size,
                              hipStream_t stream) {
    const float* x_in = (const float*)d_in[0];
    const int*   ei   = (const int*)d_in[1];
    const float* W    = (const float*)d_in[2];
    const float* Wphi = (const float*)d_in[3];
    const float* bias = (const float*)d_in[4];
    float*       out  = (float*)d_out;

    const int N = in_sizes[0] / D;
    const int E = in_sizes[1] / 2;

    // Workspace layout (floats): xw | agg | xbuf | dinv | WphiP | ATP
    float* ws    = (float*)d_ws;
    const size_t nd = (size_t)N * D;
    float* xw    = ws;
    float* agg   = ws + nd;
    float* xbuf  = ws + 2 * nd;
    float* dinv  = ws + 3 * nd;
    float* WphiP = dinv + N;
    float* ATP   = WphiP + (size_t)D * D;

    const int B = 256;

    // --- precompute (re-run every call: deterministic, depends only on inputs) ---
    init_deg_kernel<<<(N + B - 1) / B, B, 0, stream>>>(dinv, N);
    count_deg_kernel<<<(E + B - 1) / B, B, 0, stream>>>(ei, dinv, E);
    finalize_dinv_kernel<<<(N + B - 1) / B, B, 0, stream>>>(dinv, N);
    prep_weights_kernel<<<(D * D + B - 1) / B, B, 0, stream>>>(W, Wphi, WphiP, ATP);

    const int gemm_blocks = (N + TILE_M - 1) / TILE_M;      // 6250
    const int edge_blocks = (E + (B / 32) - 1) / (B / 32);  // one wave per edge
    const int comb_blocks = (N * (D / 4) + B - 1) / B;

    for (int it = 0; it < NUM_ITERS; ++it) {
        const float* xi = (it == 0) ? x_in : xbuf;
        float* xo = (it == NUM_ITERS - 1) ? out : xbuf;
        gemm_dual_kernel<<<gemm_blocks, B, 0, stream>>>(xi, WphiP, ATP, bias, xw, agg, N);
        edge_scatter_kernel<<<edge_blocks, B, 0, stream>>>(ei, xw, dinv, agg, E);
        combine_kernel<<<comb_blocks, B, 0, stream>>>(xi, xw, agg, dinv, xo, N);
    }
}